// GlobalLinear_16088947491454
// MI455X (gfx1250) — compile-verified
//
#include <hip/hip_runtime.h>
#include <hip/hip_bf16.h>

// ---------------- problem constants (from reference) ----------------
#define G_     500
#define N_     500000
#define E_     1000000
#define DN_    128
#define DE_    64
#define DG_    128
#define DO_    128
#define K_     (DN_ + DE_ + DG_)   // 320
#define GPAD_  512                 // 500 padded to 32 tiles of 16

typedef float v2f __attribute__((ext_vector_type(2)));
typedef float v8f __attribute__((ext_vector_type(8)));

// ---------------- kernel 1: exclusive prefix sums of segment lengths ----------------
__global__ void gl_prefix(const int* __restrict__ nlen, const int* __restrict__ elen,
                          int* __restrict__ offs) {
  if (threadIdx.x == 0 && blockIdx.x == 0) {
    int sn = 0, se = 0;
    for (int g = 0; g < G_; ++g) {
      offs[g]      = sn; sn += nlen[g];
      offs[G_ + g] = se; se += elen[g];
    }
  }
}

// ---------------- kernel 2: pack Acat padding/global columns + Wcat ----------------
// Acat[GPAD_][K_]: cols [0,128)=node_agg (agg kernel), [128,192)=edge_agg (agg kernel),
//                  [192,320)=global_features; rows >= G_ are zero.
// Wcat[DO_][K_]:   concat(W_node_row, W_edges_row, W_global_row) per output channel.
__global__ void gl_pack(const float* __restrict__ gfeat,
                        const float* __restrict__ Wn, const float* __restrict__ We,
                        const float* __restrict__ Wg,
                        float* __restrict__ Acat, float* __restrict__ Wcat) {
  const int NA = GPAD_ * K_;
  const int NW = DO_ * K_;
  int idx = blockIdx.x * blockDim.x + threadIdx.x;
  if (idx < NA) {
    int row = idx / K_;
    int col = idx - row * K_;
    if (row >= G_) {
      Acat[idx] = 0.0f;
    } else if (col >= DN_ + DE_) {
      Acat[idx] = gfeat[row * DG_ + (col - DN_ - DE_)];
    }
    // cols [0,192) for row < G_ are written by the aggregation kernel (disjoint)
  } else {
    int j = idx - NA;
    if (j < NW) {
      int o = j / K_;
      int k = j - o * K_;
      float v;
      if (k < DN_)            v = Wn[o * DN_ + k];
      else if (k < DN_ + DE_) v = We[o * DE_ + (k - DN_)];
      else                    v = Wg[o * DG_ + (k - DN_ - DE_)];
      Wcat[j] = v;
    }
  }
}

// ---------------- kernel 3: per-graph segment mean (nodes: y=0, edges: y=1) ----------------
// One 1024-thread block per (graph, kind). float4 coalesced streaming loads,
// deterministic LDS tree reduction (no float atomics -> bitwise-reproducible).
__global__ __launch_bounds__(1024)
void gl_agg(const float* __restrict__ nodes, const float* __restrict__ edges,
            const int* __restrict__ offs,
            const int* __restrict__ nlen, const int* __restrict__ elen,
            float* __restrict__ Acat) {
  __shared__ float4 red[1024];
  const int g   = blockIdx.x;
  const int tid = threadIdx.x;
  const bool isEdge = (blockIdx.y == 1);

  const float* src; int len, start, D4, colbase;
  if (!isEdge) { src = nodes; len = nlen[g]; start = offs[g];      D4 = DN_ / 4; colbase = 0;   }
  else         { src = edges; len = elen[g]; start = offs[G_ + g]; D4 = DE_ / 4; colbase = DN_; }

  const int cv  = tid & (D4 - 1);   // float4 column
  const int rg  = tid / D4;         // row group
  const int nrg = 1024 / D4;        // groups per block

  float4 acc = make_float4(0.f, 0.f, 0.f, 0.f);
  const float4* s4 = (const float4*)src;
  for (int r = rg; r < len; r += nrg) {
    float4 v = s4[(long)(start + r) * D4 + cv];
    acc.x += v.x; acc.y += v.y; acc.z += v.z; acc.w += v.w;
  }
  red[tid] = acc;
  __syncthreads();
  for (int s = nrg >> 1; s > 0; s >>= 1) {
    if (rg < s) {
      float4 o = red[tid + s * D4];
      acc.x += o.x; acc.y += o.y; acc.z += o.z; acc.w += o.w;
      red[tid] = acc;
    }
    __syncthreads();
  }
  if (rg == 0) {
    float inv = 1.0f / (float)max(len, 1);
    float4* dst = (float4*)(Acat + (long)g * K_ + colbase);
    dst[cv] = make_float4(acc.x * inv, acc.y * inv, acc.z * inv, acc.w * inv);
  }
}

// ---------------- kernel 4: WMMA fp32 GEMM  out = Acat (GPADxK) * Wcat^T (KxDO) + bias ----
// One wave per 16x16 output tile: block = M tile (32 blocks), wave = N tile (8 waves).
// A layout (16x4 f32): lanes 0-15 -> M=lane, v0=K0 v1=K1; lanes 16-31 -> M=lane-16, v0=K2 v1=K3.
// B layout mirrors with N; B[k][n] = Wcat[n][k].
// C/D: VGPR r, lanes 0-15 -> M = r; lanes 16-31 -> M = r + 8.
__global__ __launch_bounds__(256)
void gl_gemm(const float* __restrict__ Acat, const float* __restrict__ Wcat,
             const float* __restrict__ bias, float* __restrict__ out) {
  const int lane = threadIdx.x & 31;
  const int wave = threadIdx.x >> 5;        // 0..7  -> N tile
  const int m0   = blockIdx.x * 16;         // 0..496 step 16 (GPAD_/16 = 32 blocks)
  const int n0   = wave * 16;
  const int l16  = lane & 15;
  const int koff = (lane >> 4) << 1;        // 0 for lanes 0-15, 2 for lanes 16-31

  const float* arow = Acat + (long)(m0 + l16) * K_;
  const float* brow = Wcat + (long)(n0 + l16) * K_;

  v8f c = {0.f, 0.f, 0.f, 0.f, 0.f, 0.f, 0.f, 0.f};
  for (int k0 = 0; k0 < K_; k0 += 4) {
    v2f a = *(const v2f*)(arow + k0 + koff);   // 8B-aligned (k0%4==0, koff in {0,2})
    v2f b = *(const v2f*)(brow + k0 + koff);
    c = __builtin_amdgcn_wmma_f32_16x16x4_f32(false, a, false, b, (short)0, c,
                                              false, false);
  }

  const float bv = bias[n0 + l16];
  const int rbase = m0 + ((lane >> 4) << 3); // +8 for the upper half-wave
  #pragma unroll
  for (int r = 0; r < 8; ++r) {
    int row = rbase + r;
    if (row < G_) out[(long)row * DO_ + n0 + l16] = c[r] + bv;
  }
}

// ---------------- launch ----------------
extern "C" void kernel_launch(void* const* d_in, const int* in_sizes, int n_in,
                              void* d_out, int out_size, void* d_ws, size_t ws_size,
                              hipStream_t stream) {
  (void)in_sizes; (void)n_in; (void)out_size; (void)ws_size;

  const float* node_feat = (const float*)d_in[0];   // [N_, DN_]
  const float* edge_feat = (const float*)d_in[1];   // [E_, DE_]
  const float* glob_feat = (const float*)d_in[2];   // [G_, DG_]
  const float* W_node    = (const float*)d_in[3];   // [DO_, DN_]
  const float* W_edges   = (const float*)d_in[4];   // [DO_, DE_]
  const float* W_global  = (const float*)d_in[5];   // [DO_, DG_]
  const float* bias      = (const float*)d_in[6];   // [DO_]
  const int*   nlen      = (const int*)d_in[7];     // [G_]
  const int*   elen      = (const int*)d_in[8];     // [G_]
  float*       out       = (float*)d_out;           // [G_, DO_]

  // workspace layout (f32): Acat[GPAD_*K_] | Wcat[DO_*K_] | offs[2*G_] ints
  float* Acat = (float*)d_ws;
  float* Wcat = Acat + (size_t)GPAD_ * K_;
  int*   offs = (int*)(Wcat + (size_t)DO_ * K_);

  // 1) prefix sums of segment lengths
  gl_prefix<<<1, 64, 0, stream>>>(nlen, elen, offs);

  // 2) pack padding rows + global columns + concatenated weights
  {
    int total  = GPAD_ * K_ + DO_ * K_;           // 204800
    int blocks = (total + 255) / 256;
    gl_pack<<<blocks, 256, 0, stream>>>(glob_feat, W_node, W_edges, W_global, Acat, Wcat);
  }

  // 3) segment means (bandwidth-dominated: ~512 MB streamed)
  gl_agg<<<dim3(G_, 2), 1024, 0, stream>>>(node_feat, edge_feat, offs, nlen, elen, Acat);

  // 4) fp32 WMMA projection + bias
  gl_gemm<<<GPAD_ / 16, 256, 0, stream>>>(Acat, Wcat, bias, out);
}